// AddRadiusEdgeIndex_73478300500255
// MI455X (gfx1250) — compile-verified
//
#include <hip/hip_runtime.h>

typedef __attribute__((ext_vector_type(2))) float v2f;
typedef __attribute__((ext_vector_type(8))) float v8f;

#define NPTS 8192
#define R2   9.0f   // RADIUS^2

// D = A(16x4) x B(4x16) + C  via V_WMMA_F32_16X16X4_F32.
// A row i   = (-2x_i, -2y_i, -2z_i, sq_i)   -> contributes sq_i - 2*dot
// B col j   = ( x_j,   y_j,   z_j,  1   )
// then add sq_j per lane (column term; column == lane&15 for all D VGPRs).
//
// Fragment layout (CDNA5 ISA 7.12.2, wave32):
//   A 16x4 f32:  lane l, vgpr r  holds A(m = l&15, k = r + 2*(l>>4))
//   B 4x16 f32:  lane l, vgpr r  holds B(k = r + 2*(l>>4), n = l&15)
//   D 16x16 f32: lane l, vgpr r  holds D(m = r + 8*(l>>4), n = l&15)
__global__ __launch_bounds__(256) void radius_mask_wmma_kernel(
    const float* __restrict__ pos, float* __restrict__ out) {
  const int lane = threadIdx.x & 31;
  const int wave = threadIdx.x >> 5;
  const int half = lane >> 4;   // 0: lanes 0-15, 1: lanes 16-31
  const int l16  = lane & 15;

  const int i0 = blockIdx.y * 128 + wave * 16;  // 8 waves -> 128 rows per block
  const int j0 = blockIdx.x * 64;               // 4 j-tiles -> 64 cols per wave

  // ---- A fragment (rows i0..i0+15, augmented K=4) ----
  const int ip = i0 + l16;
  const float ax = pos[3 * ip + 0];
  const float ay = pos[3 * ip + 1];
  const float az = pos[3 * ip + 2];
  const float sqi = ax * ax + ay * ay + az * az;

  v2f afrag;
  afrag[0] = half ? (-2.0f * az) : (-2.0f * ax);  // k = 0 or 2
  afrag[1] = half ? sqi          : (-2.0f * ay);  // k = 1 or 3

  const int rowbase = i0 + half * 8;

#pragma unroll
  for (int t = 0; t < 4; ++t) {
    const int jp = j0 + t * 16 + l16;
    const float bx = pos[3 * jp + 0];
    const float by = pos[3 * jp + 1];
    const float bz = pos[3 * jp + 2];
    const float sqj = bx * bx + by * by + bz * bz;

    v2f bfrag;
    bfrag[0] = half ? bz   : bx;   // k = 0 or 2
    bfrag[1] = half ? 1.0f : by;   // k = 1 or 3

    v8f c = {};
    // 8 args: (neg_a, A, neg_b, B, c_mod, C, reuse_a, reuse_b)
    v8f d = __builtin_amdgcn_wmma_f32_16x16x4_f32(
        false, afrag, false, bfrag, (short)0, c, false, false);

#pragma unroll
    for (int r = 0; r < 8; ++r) {
      // d2 = sq_i + sq_j - 2*dot ; clamp-to-0 is a no-op for the <= 9 test
      const float d2 = d[r] + sqj;
      out[(size_t)(rowbase + r) * NPTS + jp] = (d2 <= R2) ? 1.0f : 0.0f;
    }
  }
}

extern "C" void kernel_launch(void* const* d_in, const int* in_sizes, int n_in,
                              void* d_out, int out_size, void* d_ws, size_t ws_size,
                              hipStream_t stream) {
  (void)in_sizes; (void)n_in; (void)out_size; (void)d_ws; (void)ws_size;
  const float* pos = (const float*)d_in[0];   // [8192, 3] f32
  float* out = (float*)d_out;                 // [8192, 8192] mask as 0.0/1.0

  dim3 grid(NPTS / 64, NPTS / 128);  // (128, 64) blocks
  dim3 block(256);                   // 8 wave32s
  radius_mask_wmma_kernel<<<grid, block, 0, stream>>>(pos, out);
}